// MyModel_56770877719159
// MI455X (gfx1250) — compile-verified
//
#include <hip/hip_runtime.h>

typedef __attribute__((ext_vector_type(2))) float v2f;
typedef __attribute__((ext_vector_type(8))) float v8f;

#define N_NODES 50000
#define N_EDGES 600000
#define D 128
#define R 8
#define MT 5            // 16-row tiles per wave; 3125 tiles = 5 * 625 blocks exactly

// ---------------------------------------------------------------------------
// h0[i] = emb[node_ids[i]]  (vectorized float4; D/4 = 32 float4 per row)
// ---------------------------------------------------------------------------
__global__ void rgcn_gather(const int* __restrict__ ids,
                            const float* __restrict__ emb,
                            float* __restrict__ h0, int total4) {
    int i = blockIdx.x * blockDim.x + threadIdx.x;
    if (i < total4) {
        int node = i >> 5;          // / (D/4)
        int d4   = i & 31;          // % (D/4)
        reinterpret_cast<float4*>(h0)[i] =
            reinterpret_cast<const float4*>(emb)[(size_t)ids[node] * 32 + d4];
    }
}

// ---------------------------------------------------------------------------
// Edge scatter (aggregate-then-transform): acc[etype][dst] += h[src]
// Each lane loads indices for a distinct edge (coalesced), then the wave
// processes the 32 edges with __shfl broadcast (ds_bpermute on wave32).
// Per edge: one b128 load of h[src] per lane + 4 global_atomic_add_f32.
// h is 25.6 MB -> L2 resident; atomics stream into the 205 MB acc.
// N_EDGES % 32 == 0 and all strides are multiples of 32 -> no bounds checks.
// ---------------------------------------------------------------------------
__global__ void rgcn_edge_scatter(const int* __restrict__ src,
                                  const int* __restrict__ dst,
                                  const int* __restrict__ etype,
                                  const float* __restrict__ hin,
                                  float* __restrict__ acc) {
    const int wave   = (blockIdx.x * blockDim.x + threadIdx.x) >> 5;
    const int lane   = threadIdx.x & 31;
    const int nwaves = (gridDim.x * blockDim.x) >> 5;

    for (int base = wave * 32; base < N_EDGES; base += nwaves * 32) {
        const int e = base + lane;          // always < N_EDGES
        const int s  = src[e];
        const int dt = dst[e];
        const int r  = etype[e];
        #pragma unroll 4
        for (int j = 0; j < 32; ++j) {
            const int sj = __shfl(s,  j);
            const int dj = __shfl(dt, j);
            const int rj = __shfl(r,  j);
            const float4 v =
                reinterpret_cast<const float4*>(hin + (size_t)sj * D)[lane];
            float* ap = acc + ((size_t)rj * N_NODES + (size_t)dj) * D + 4 * lane;
            atomicAdd(ap + 0, v.x);
            atomicAdd(ap + 1, v.y);
            atomicAdd(ap + 2, v.z);
            atomicAdd(ap + 3, v.w);
        }
    }
}

// ---------------------------------------------------------------------------
// Fused GEMM: hout = hin*Wself + sum_r acc_r*W_r + bias
// One wave owns MT=5 16x16 output tiles in the same column strip; B (the W
// tile) is loaded once per k-step and reused across the 5 WMMAs, cutting the
// load/WMMA ratio from 3.0 to 1.4. Block = 8 waves = all 8 column tiles.
//
// Per-lane VGPR layouts (ISA 7.12.2, 32-bit data):
//   A (16x4):  row = lane&15, koff = (lane>>4)*2, holds k+koff, k+koff+1
//   B (4x16):  col = lane&15, koff = (lane>>4)*2, holds rows k+koff, k+koff+1
//   C/D:       vgpr g -> row g + 8*(lane>=16), col = lane&15
// ---------------------------------------------------------------------------
__global__ void rgcn_gemm(const float* __restrict__ hin,
                          const float* __restrict__ acc,
                          const float* __restrict__ W,      // [R, D, D]
                          const float* __restrict__ Wself,  // [D, D]
                          const float* __restrict__ bias,   // [D]
                          float* __restrict__ hout) {
    const int wave    = threadIdx.x >> 5;      // 0..7 -> column tile
    const int lane    = threadIdx.x & 31;
    const int colbase = wave * 16;
    const int row     = lane & 15;
    const int col     = lane & 15;
    const int koff    = (lane >> 4) * 2;
    const int rowbase = blockIdx.x * (MT * 16);

    v8f c[MT];
    #pragma unroll
    for (int t = 0; t < MT; ++t)
        c[t] = (v8f){0.f, 0.f, 0.f, 0.f, 0.f, 0.f, 0.f, 0.f};

    #pragma unroll 1
    for (int m = 0; m < R + 1; ++m) {
        const float* In = (m == 0) ? hin
                                   : acc + (size_t)(m - 1) * N_NODES * D;
        const float* Wm = (m == 0) ? Wself
                                   : W + (size_t)(m - 1) * D * D;
        const float* arow = In + (size_t)(rowbase + row) * D + koff;
        const float* bcol = Wm + (size_t)koff * D + colbase + col;
        #pragma unroll 2
        for (int k = 0; k < D; k += 4) {
            v2f b;
            b.x = bcol[(size_t)k * D];           // (k+koff,   col)
            b.y = bcol[(size_t)(k + 1) * D];     // (k+koff+1, col)
            #pragma unroll
            for (int t = 0; t < MT; ++t) {
                v2f a;
                a.x = arow[(size_t)t * 16 * D + k];       // (row+16t, k+koff)
                a.y = arow[(size_t)t * 16 * D + k + 1];   // (row+16t, k+koff+1)
                c[t] = __builtin_amdgcn_wmma_f32_16x16x4_f32(
                        /*neg_a=*/false, a, /*neg_b=*/false, b,
                        /*c_mod=*/(short)0, c[t],
                        /*reuse_a=*/false, /*reuse_b=*/false);
            }
        }
    }

    const float bv  = bias[colbase + col];
    const int rhalf = (lane >> 4) * 8;
    #pragma unroll
    for (int t = 0; t < MT; ++t) {
        #pragma unroll
        for (int g = 0; g < 8; ++g) {
            const int rr = rowbase + t * 16 + g + rhalf;
            hout[(size_t)rr * D + colbase + col] = c[t][g] + bv;
        }
    }
}

// ---------------------------------------------------------------------------
extern "C" void kernel_launch(void* const* d_in, const int* in_sizes, int n_in,
                              void* d_out, int out_size, void* d_ws, size_t ws_size,
                              hipStream_t stream) {
    const int*   node_ids = (const int*)d_in[0];
    const int*   src      = (const int*)d_in[1];
    const int*   dst      = (const int*)d_in[2];
    const int*   etype    = (const int*)d_in[3];
    const float* emb      = (const float*)d_in[4];
    const float* W1       = (const float*)d_in[5];
    const float* Ws1      = (const float*)d_in[6];
    const float* b1       = (const float*)d_in[7];
    const float* W2       = (const float*)d_in[8];
    const float* Ws2      = (const float*)d_in[9];
    const float* b2       = (const float*)d_in[10];

    float* acc = (float*)d_ws;                                // R*N*D floats
    float* h0  = acc + (size_t)R * N_NODES * D;               // N*D floats
    float* h1  = h0 + (size_t)N_NODES * D;                    // N*D floats
    float* h2  = (float*)d_out;

    const size_t accBytes = (size_t)R * N_NODES * D * sizeof(float);
    const int    gemmGrid = (N_NODES / 16) / MT;              // 625

    // h0 = emb[node_ids]
    {
        int total4 = N_NODES * D / 4;
        rgcn_gather<<<(total4 + 255) / 256, 256, 0, stream>>>(node_ids, emb, h0, total4);
    }

    // ---- layer 1 ----
    hipMemsetAsync(acc, 0, accBytes, stream);
    rgcn_edge_scatter<<<1024, 256, 0, stream>>>(src, dst, etype, h0, acc);
    rgcn_gemm<<<gemmGrid, 256, 0, stream>>>(h0, acc, W1, Ws1, b1, h1);

    // ---- layer 2 ----
    hipMemsetAsync(acc, 0, accBytes, stream);
    rgcn_edge_scatter<<<1024, 256, 0, stream>>>(src, dst, etype, h1, acc);
    rgcn_gemm<<<gemmGrid, 256, 0, stream>>>(h1, acc, W2, Ws2, b2, h2);
}